// CGIB_78520592106195
// MI455X (gfx1250) — compile-verified
//
#include <hip/hip_runtime.h>
#include <math.h>

typedef __attribute__((ext_vector_type(2))) float v2f;
typedef __attribute__((ext_vector_type(8))) float v8f;

#define NUQ   8192      // solute nodes
#define NVQ   8192      // solvent nodes
#define NGB   256       // num graphs
#define HD    300       // hidden dim H
#define DD    600       // D = 2H
#define X2S   600       // row stride of su2/sv2 ([su | prime])
#define QSD   1200      // q_star dim = 2D
#define G4D   2400      // gates dim = 4D
#define MAPN  8192      // interaction map is [8192, 8192]

__device__ __forceinline__ v8f wmma4(v2f a, v2f b, v8f c) {
  // D = A(16x4,f32) * B(4x16,f32) + C(16x16,f32)
  return __builtin_amdgcn_wmma_f32_16x16x4_f32(false, a, false, b, (short)0, c,
                                               false, false);
}

__device__ __forceinline__ v2f ld2(const float* p) {
  return *(const v2f*)p;   // caller guarantees 8B alignment
}

// ---------------------------------------------------------------- fills
__global__ void fill4_kernel(float4* __restrict__ p, size_t n4) {
  size_t i = (size_t)blockIdx.x * blockDim.x + threadIdx.x;
  size_t stride = (size_t)gridDim.x * blockDim.x;
  float4 z; z.x = 0.f; z.y = 0.f; z.z = 0.f; z.w = 0.f;
  for (; i < n4; i += stride) p[i] = z;
}

// ---------------------------------------------------------------- l2 norm
// one wave per row; rows [0,NUQ) -> su2, rows [NUQ, NUQ+NVQ) -> sv2
__global__ void norm_kernel(const float* __restrict__ sx,
                            const float* __restrict__ vx,
                            float* __restrict__ su2, float* __restrict__ sv2) {
  int wave = (int)((blockIdx.x * blockDim.x + threadIdx.x) >> 5);
  int lane = threadIdx.x & 31;
  const float* src;
  float* dst;
  if (wave < NUQ) {
    src = sx + (size_t)wave * HD;
    dst = su2 + (size_t)wave * X2S;
  } else if (wave < NUQ + NVQ) {
    int r = wave - NUQ;
    src = vx + (size_t)r * HD;
    dst = sv2 + (size_t)r * X2S;
  } else {
    return;
  }
  float ss = 0.f;
  for (int k = lane; k < HD; k += 32) { float v = src[k]; ss += v * v; }
  for (int off = 16; off > 0; off >>= 1) ss += __shfl_xor(ss, off, 32);
  float scale = 1.0f / fmaxf(sqrtf(ss), 1e-12f);
  for (int k = lane; k < HD; k += 32) dst[k] = src[k] * scale;
}

// ---------------------------------------------------------------- graph bounds
__global__ void bounds_kernel(const int* __restrict__ ub, const int* __restrict__ vb,
                              int* __restrict__ us, int* __restrict__ ue,
                              int* __restrict__ vs, int* __restrict__ ve) {
  int t = blockIdx.x * blockDim.x + threadIdx.x;
  if (t >= 2 * NGB) return;
  int g = t & (NGB - 1);
  const int* b = (t < NGB) ? ub : vb;
  int n = (t < NGB) ? NUQ : NVQ;
  int lo = 0, hi = n;
  while (lo < hi) { int mid = (lo + hi) >> 1; if (b[mid] < g) lo = mid + 1; else hi = mid; }
  int start = lo;
  hi = n;
  while (lo < hi) { int mid = (lo + hi) >> 1; if (b[mid] <= g) lo = mid + 1; else hi = mid; }
  if (t < NGB) { us[g] = start; ue[g] = lo; } else { vs[g] = start; ve[g] = lo; }
}

// ---------------------------------------------------------------- interaction
// one block per graph; waves tile the [nu x nv] rectangle with 16x16 WMMA tiles.
// Edge rows/cols are CLAMPED (not zeroed): padding lanes compute garbage that is
// never stored, so the K loop has zero predication -> clean b64 loads + WMMA.
__global__ void interact_kernel(const float* __restrict__ su2,
                                const float* __restrict__ sv2,
                                const int* __restrict__ us_, const int* __restrict__ ue_,
                                const int* __restrict__ vs_, const int* __restrict__ ve_,
                                float* __restrict__ map) {
  int g = blockIdx.x;
  int us = us_[g], ue = ue_[g], vs = vs_[g], ve = ve_[g];
  int nu = ue - us, nv = ve - vs;
  if (nu <= 0 || nv <= 0) return;
  int ntm = (nu + 15) >> 4, ntn = (nv + 15) >> 4;
  int wave = threadIdx.x >> 5, nwaves = blockDim.x >> 5;
  int lane = threadIdx.x & 31;
  int half = lane >> 4, lm = lane & 15;
  int koff = half * 2;
  for (int t = wave; t < ntm * ntn; t += nwaves) {
    int tm = t / ntn, tn = t - tm * ntn;
    int arow = min(us + tm * 16 + lm, ue - 1);   // clamped A row (M)
    int bcol = min(vs + tn * 16 + lm, ve - 1);   // clamped B col (N)
    const float* ap = su2 + (size_t)arow * X2S + koff;
    const float* bp = sv2 + (size_t)bcol * X2S + koff;
    v8f acc = {};
    int k0 = 0;
    for (; k0 + 8 <= HD; k0 += 8) {     // 296 of 300
      v2f a0 = ld2(ap + k0);
      v2f b0 = ld2(bp + k0);
      v2f a1 = ld2(ap + k0 + 4);
      v2f b1 = ld2(bp + k0 + 4);
      acc = wmma4(a0, b0, acc);
      acc = wmma4(a1, b1, acc);
    }
    for (; k0 < HD; k0 += 4) {          // tail (k0 = 296)
      v2f a = ld2(ap + k0);
      v2f b = ld2(bp + k0);
      acc = wmma4(a, b, acc);
    }
    int col = vs + tn * 16 + lm;
    if (col < ve) {
#pragma unroll
      for (int i = 0; i < 8; i++) {
        int row = us + tm * 16 + half * 8 + i;
        if (row < ue) map[(size_t)row * MAPN + col] = acc[i];
      }
    }
  }
}

// ---------------------------------------------------------------- primes
// blockIdx.y==0: solute_prime = C_g @ sv_g    -> su2[:,300:600]
// blockIdx.y==1: solvent_prime = C_g^T @ su_g -> sv2[:,300:600]
// M/N edges clamped (padding never stored); K tail handled with value selects
// on clamped (safe) addresses -> no exec-mask diamonds in the hot loop.
__global__ void prime_kernel(const float* __restrict__ map,
                             float* __restrict__ su2, float* __restrict__ sv2,
                             const int* __restrict__ us_, const int* __restrict__ ue_,
                             const int* __restrict__ vs_, const int* __restrict__ ve_) {
  int g = blockIdx.x;
  int side = blockIdx.y;
  int us = us_[g], ue = ue_[g], vs = vs_[g], ve = ve_[g];
  int mBase = (side == 0) ? us : vs;
  int mEnd  = (side == 0) ? ue : ve;
  int kBase = (side == 0) ? vs : us;
  int kEnd  = (side == 0) ? ve : ue;
  int nm = mEnd - mBase, nk = kEnd - kBase;
  if (nm <= 0) return;
  const float* xsrc = (side == 0) ? sv2 : su2;
  float* xdst = (side == 0) ? su2 : sv2;
  int ntm = (nm + 15) >> 4;
  const int ntn = (HD + 15) / 16;  // 19
  int wave = threadIdx.x >> 5, nwaves = blockDim.x >> 5;
  int lane = threadIdx.x & 31;
  int half = lane >> 4, lm = lane & 15;
  int koff = half * 2;
  int nk4 = nk & ~3;               // unconditional part of K
  for (int t = wave; t < ntm * ntn; t += nwaves) {
    int tm = t / ntn, tn = t - tm * ntn;
    int mrow = min(mBase + tm * 16 + lm, mEnd - 1);   // clamped
    int ncol = tn * 16 + lm;
    int ncolc = min(ncol, HD - 1);                    // clamped
    v8f acc = {};
    if (side == 0) {
      // A[m][k] = C[mrow][kBase+k] (contiguous in k, but kBase parity unknown
      // -> use b32 pairs). B[k][n] = xsrc[kBase+k][ncol] (row-strided b32).
      const float* mp = map + (size_t)mrow * MAPN + kBase + koff;
      const float* bp = xsrc + (size_t)kBase * X2S + ncolc;
      for (int k0 = 0; k0 < nk4; k0 += 4) {
        v2f a, b;
        a.x = mp[k0];
        a.y = mp[k0 + 1];
        b.x = bp[(size_t)(k0 + koff) * X2S];
        b.y = bp[(size_t)(k0 + koff + 1) * X2S];
        acc = wmma4(a, b, acc);
      }
    } else {
      // A[m][k] = C[kBase+k][mrow] (column of C, stride MAPN)
      const float* mp = map + (size_t)(kBase + koff) * MAPN + mrow;
      const float* bp = xsrc + (size_t)kBase * X2S + ncolc;
      for (int k0 = 0; k0 < nk4; k0 += 4) {
        v2f a, b;
        a.x = mp[(size_t)k0 * MAPN];
        a.y = mp[(size_t)(k0 + 1) * MAPN];
        b.x = bp[(size_t)(k0 + koff) * X2S];
        b.y = bp[(size_t)(k0 + koff + 1) * X2S];
        acc = wmma4(a, b, acc);
      }
    }
    if (nk4 < nk) {                 // K tail: safe clamped loads + selects
      int k0e = nk4 + koff;
      int kc0 = min(k0e, nk - 1);
      int kc1 = min(k0e + 1, nk - 1);
      v2f a, b;
      float ax, ay;
      if (side == 0) {
        const float* mp = map + (size_t)mrow * MAPN + kBase;
        ax = mp[kc0];
        ay = mp[kc1];
      } else {
        ax = map[(size_t)(kBase + kc0) * MAPN + mrow];
        ay = map[(size_t)(kBase + kc1) * MAPN + mrow];
      }
      float bx = xsrc[(size_t)(kBase + kc0) * X2S + ncolc];
      float by = xsrc[(size_t)(kBase + kc1) * X2S + ncolc];
      a.x = (k0e < nk) ? ax : 0.f;
      a.y = (k0e + 1 < nk) ? ay : 0.f;
      b.x = (k0e < nk) ? bx : 0.f;
      b.y = (k0e + 1 < nk) ? by : 0.f;
      acc = wmma4(a, b, acc);
    }
    if (ncol < HD) {
#pragma unroll
      for (int i = 0; i < 8; i++) {
        int m = mBase + tm * 16 + half * 8 + i;
        if (m < mEnd) xdst[(size_t)m * X2S + HD + ncol] = acc[i];
      }
    }
  }
}

// ---------------------------------------------------------------- LSTM gates
// gates[256,2400] = q_star[256,1200] @ W_ih^T + h[256,600] @ W_hh^T + b_ih + b_hh
// All dims are multiples of 16/8 -> fully unconditional b64 loads.
__global__ void gates_kernel(const float* __restrict__ qs, const float* __restrict__ h,
                             const float* __restrict__ Wih, const float* __restrict__ Whh,
                             const float* __restrict__ bih, const float* __restrict__ bhh,
                             float* __restrict__ gates) {
  const int NTN = G4D / 16;  // 150
  const int NTM = NGB / 16;  // 16
  int wg = (int)((blockIdx.x * blockDim.x + threadIdx.x) >> 5);
  if (wg >= NTM * NTN) return;
  int tm = wg / NTN, tn = wg - tm * NTN;
  int lane = threadIdx.x & 31;
  int half = lane >> 4, lm = lane & 15;
  int koff = half * 2;
  int arow = tm * 16 + lm;   // batch row (<256)
  int brow = tn * 16 + lm;   // gate index (<2400); B[k][n] = W[brow][k]
  v8f acc = {};
  {
    const float* ap = qs + (size_t)arow * QSD + koff;
    const float* bp = Wih + (size_t)brow * QSD + koff;
    for (int k0 = 0; k0 < QSD; k0 += 8) {
      v2f a0 = ld2(ap + k0);
      v2f b0 = ld2(bp + k0);
      v2f a1 = ld2(ap + k0 + 4);
      v2f b1 = ld2(bp + k0 + 4);
      acc = wmma4(a0, b0, acc);
      acc = wmma4(a1, b1, acc);
    }
  }
  {
    const float* ap = h + (size_t)arow * DD + koff;
    const float* bp = Whh + (size_t)brow * DD + koff;
    for (int k0 = 0; k0 < DD; k0 += 8) {
      v2f a0 = ld2(ap + k0);
      v2f b0 = ld2(bp + k0);
      v2f a1 = ld2(ap + k0 + 4);
      v2f b1 = ld2(bp + k0 + 4);
      acc = wmma4(a0, b0, acc);
      acc = wmma4(a1, b1, acc);
    }
  }
  int n = tn * 16 + lm;
  float bias = bih[n] + bhh[n];
#pragma unroll
  for (int i = 0; i < 8; i++) {
    int m = tm * 16 + half * 8 + i;
    gates[(size_t)m * G4D + n] = acc[i] + bias;
  }
}

// ---------------------------------------------------------------- LSTM cell
__global__ void lstm_cell_kernel(const float* __restrict__ gates,
                                 float* __restrict__ h, float* __restrict__ c,
                                 float* __restrict__ qs) {
  int t = blockIdx.x * blockDim.x + threadIdx.x;
  if (t >= NGB * DD) return;
  int b = t / DD, j = t - b * DD;
  const float* gr = gates + (size_t)b * G4D;
  float ig = gr[j], fg = gr[DD + j], gg = gr[2 * DD + j], og = gr[3 * DD + j];
  float si = 1.f / (1.f + expf(-ig));
  float sf = 1.f / (1.f + expf(-fg));
  float so = 1.f / (1.f + expf(-og));
  float cn = sf * c[t] + si * tanhf(gg);
  float hn = so * tanhf(cn);
  c[t] = cn;
  h[t] = hn;
  qs[(size_t)b * QSD + j] = hn;  // q half of next q_star
}

// ---------------------------------------------------------------- attention
__global__ void attn_kernel(const float* __restrict__ x2, const float* __restrict__ h,
                            const int* __restrict__ s_, const int* __restrict__ e_,
                            float* __restrict__ ebuf, float* __restrict__ qs) {
  int g = blockIdx.x;
  int ns = s_[g], ne = e_[g];
  __shared__ float sred[8];
  __shared__ float s_emax, s_asum;
  int wave = threadIdx.x >> 5, lane = threadIdx.x & 31;
  if (threadIdx.x == 0) s_asum = 0.f;
  const float* hg = h + (size_t)g * DD;
  float lmax = -3.402823466e+38f;
  for (int i = ns + wave; i < ne; i += 8) {
    const float* xr = x2 + (size_t)i * X2S;
    float dot = 0.f;
    for (int k = lane; k < DD; k += 32) dot += xr[k] * hg[k];
    for (int off = 16; off > 0; off >>= 1) dot += __shfl_xor(dot, off, 32);
    if (lane == 0) ebuf[i] = dot;
    lmax = fmaxf(lmax, dot);
  }
  if (lane == 0) sred[wave] = lmax;
  __threadfence_block();
  __syncthreads();
  if (threadIdx.x == 0) {
    float m = -3.402823466e+38f;
    for (int w = 0; w < 8; w++) m = fmaxf(m, sred[w]);
    s_emax = m;
  }
  __syncthreads();
  float emax = s_emax;
  float lsum = 0.f;
  for (int i = ns + (int)threadIdx.x; i < ne; i += (int)blockDim.x)
    lsum += expf(ebuf[i] - emax);
  for (int off = 16; off > 0; off >>= 1) lsum += __shfl_xor(lsum, off, 32);
  if (lane == 0) atomicAdd(&s_asum, lsum);
  __syncthreads();
  float denom = s_asum + 1e-16f;
  for (int hc = threadIdx.x; hc < DD; hc += blockDim.x) {
    float r = 0.f;
    for (int i = ns; i < ne; i++)
      r += expf(ebuf[i] - emax) * x2[(size_t)i * X2S + hc];
    qs[(size_t)g * QSD + DD + hc] = r / denom;
  }
}

// ---------------------------------------------------------------- prediction
__global__ void predict_kernel(const float* __restrict__ qsU, const float* __restrict__ qsV,
                               const float* __restrict__ pW, const float* __restrict__ pb,
                               float* __restrict__ out) {
  int wave = (int)((blockIdx.x * blockDim.x + threadIdx.x) >> 5);
  int lane = threadIdx.x & 31;
  if (wave >= NGB) return;
  const float* qu = qsU + (size_t)wave * QSD;
  const float* qv = qsV + (size_t)wave * QSD;
  float s = 0.f;
  for (int k = lane; k < QSD; k += 32) s += qu[k] * pW[k] + qv[k] * pW[QSD + k];
  for (int off = 16; off > 0; off >>= 1) s += __shfl_xor(s, off, 32);
  if (lane == 0) out[wave] = s + pb[0];
}

// ---------------------------------------------------------------- launch
extern "C" void kernel_launch(void* const* d_in, const int* in_sizes, int n_in,
                              void* d_out, int out_size, void* d_ws, size_t ws_size,
                              hipStream_t stream) {
  const float* solute_x  = (const float*)d_in[0];
  const float* solvent_x = (const float*)d_in[1];
  const int*   ubatch    = (const int*)d_in[2];
  const int*   vbatch    = (const int*)d_in[3];
  // d_in[4] = num_graphs (hardcoded NGB)
  const float* Wih = (const float*)d_in[5];
  const float* Whh = (const float*)d_in[6];
  const float* bih = (const float*)d_in[7];
  const float* bhh = (const float*)d_in[8];
  const float* pW  = (const float*)d_in[9];
  const float* pb  = (const float*)d_in[10];

  float* out_pred = (float*)d_out;          // [256]
  float* out_map  = out_pred + NGB;         // [8192*8192] row-major

  // workspace layout (all chunks multiples of 4 floats)
  float* w = (float*)d_ws;
  float* su2 = w;              w += (size_t)NUQ * X2S;   // [8192,600]
  float* sv2 = w;              w += (size_t)NVQ * X2S;
  float* qsU = w;              w += (size_t)NGB * QSD;
  float* hU  = w;              w += (size_t)NGB * DD;
  float* cU  = w;              w += (size_t)NGB * DD;
  float* qsV = w;              w += (size_t)NGB * QSD;
  float* hV  = w;              w += (size_t)NGB * DD;
  float* cV  = w;              w += (size_t)NGB * DD;
  float* gtU = w;              w += (size_t)NGB * G4D;
  float* gtV = w;              w += (size_t)NGB * G4D;
  float* eU  = w;              w += NUQ;
  float* eV  = w;              w += NVQ;
  int* ib = (int*)w;
  int* ius = ib;           int* iue = ib + NGB;
  int* ivs = ib + 2 * NGB; int* ive = ib + 3 * NGB;

  // 1) zero interaction map region of d_out and set2set state (qsU..cV)
  fill4_kernel<<<8192, 256, 0, stream>>>((float4*)out_map,
                                         (size_t)MAPN * MAPN / 4);
  size_t stateFloats = (size_t)2 * (NGB * QSD + 2 * NGB * DD);
  fill4_kernel<<<512, 256, 0, stream>>>((float4*)qsU, stateFloats / 4);

  // 2) l2-normalize -> first 300 cols of su2/sv2
  norm_kernel<<<(NUQ + NVQ) / 8, 256, 0, stream>>>(solute_x, solvent_x, su2, sv2);

  // 3) per-graph node ranges
  bounds_kernel<<<2, 256, 0, stream>>>(ubatch, vbatch, ius, iue, ivs, ive);

  // 4) block-diagonal interaction map via fp32 WMMA
  interact_kernel<<<NGB, 256, 0, stream>>>(su2, sv2, ius, iue, ivs, ive, out_map);

  // 5) solute_prime / solvent_prime via fp32 WMMA (cols 300..599)
  prime_kernel<<<dim3(NGB, 2), 256, 0, stream>>>(out_map, su2, sv2,
                                                 ius, iue, ivs, ive);

  // 6) set2set, 2 steps each side
  const float* x2s[2] = { su2, sv2 };
  float* qss[2] = { qsU, qsV };
  float* hs[2]  = { hU, hV };
  float* cs[2]  = { cU, cV };
  float* gts[2] = { gtU, gtV };
  float* ebs[2] = { eU, eV };
  const int* ss[2] = { ius, ivs };
  const int* es[2] = { iue, ive };
  for (int side = 0; side < 2; ++side) {
    for (int step = 0; step < 2; ++step) {
      gates_kernel<<<300, 256, 0, stream>>>(qss[side], hs[side], Wih, Whh,
                                            bih, bhh, gts[side]);
      lstm_cell_kernel<<<(NGB * DD + 255) / 256, 256, 0, stream>>>(
          gts[side], hs[side], cs[side], qss[side]);
      attn_kernel<<<NGB, 256, 0, stream>>>(x2s[side], hs[side], ss[side],
                                           es[side], ebs[side], qss[side]);
    }
  }

  // 7) predictions
  predict_kernel<<<NGB / 8, 256, 0, stream>>>(qsU, qsV, pW, pb, out_pred);
}